// LSAttention_73985106641231
// MI455X (gfx1250) — compile-verified
//
#include <hip/hip_runtime.h>
#include <math.h>

typedef __attribute__((ext_vector_type(2))) float v2f;
typedef __attribute__((ext_vector_type(8))) float v8f;

#define B_   64
#define T_   2000
#define E_   512
#define R_   1024
#define A_   128
#define NF_  32
#define KW_  31
#define PAD_ 15

// ---------------------------------------------------------------------------
// CDNA5 async global->LDS copy (VGLOBAL opcode 98, tracked on ASYNCcnt).
// VDST operand = VGPR holding LDS byte offset; VADDR = 64-bit global address.
// ---------------------------------------------------------------------------
__device__ __forceinline__ void async_load_b128(unsigned lds_off,
                                                const void* gptr) {
  unsigned long long ga = (unsigned long long)gptr;
  asm volatile("global_load_async_to_lds_b128 %0, %1, off"
               :: "v"(lds_off), "v"(ga)
               : "memory");
}
__device__ __forceinline__ void wait_asynccnt0() {
  asm volatile("s_wait_asynccnt 0" ::: "memory");
}

// ---------------------------------------------------------------------------
// Kernel 1: pq[b][a] = sum_r query[b][r] * W_query[a][r]
// M=64, N=128, K=1024 via V_WMMA_F32_16X16X4_F32. One wave per 16x16 tile.
// ---------------------------------------------------------------------------
__global__ __launch_bounds__(32) void pq_wmma_kernel(
    const float* __restrict__ query, const float* __restrict__ Wq,
    float* __restrict__ pq) {
  const int m0   = blockIdx.x * 16;
  const int n0   = blockIdx.y * 16;
  const int lane = threadIdx.x;
  const int half = lane >> 4;
  const int idx  = lane & 15;
  v8f c = {};
  const float* aptr = query + (size_t)(m0 + idx) * R_ + 2 * half;
  const float* bptr = Wq    + (size_t)(n0 + idx) * R_ + 2 * half;
  for (int k0 = 0; k0 < R_; k0 += 4) {
    v2f a = *(const v2f*)(aptr + k0);
    v2f b = *(const v2f*)(bptr + k0);
    c = __builtin_amdgcn_wmma_f32_16x16x4_f32(false, a, false, b,
                                              (short)0, c, false, false);
  }
#pragma unroll
  for (int i = 0; i < 8; ++i)
    pq[(size_t)(m0 + i + 8 * half) * A_ + n0 + idx] = c[i];
}

// ---------------------------------------------------------------------------
// Kernel 2: loc[b][f][t] = sum_{c,k} conv_w[f][c][k] * aw[b][c][t+k-15]
// ---------------------------------------------------------------------------
__global__ __launch_bounds__(256) void conv_kernel(
    const float* __restrict__ aw, const float* __restrict__ cw,
    float* __restrict__ loc) {
  const int b = blockIdx.x;
  const int f = blockIdx.y;
  const int t = blockIdx.z * 256 + threadIdx.x;
  if (t >= T_) return;
  const float* a0 = aw + (size_t)(b * 2 + 0) * T_;
  const float* a1 = aw + (size_t)(b * 2 + 1) * T_;
  const float* w0 = cw + (size_t)(f * 2 + 0) * KW_;
  const float* w1 = cw + (size_t)(f * 2 + 1) * KW_;
  float acc = 0.f;
#pragma unroll
  for (int k = 0; k < KW_; ++k) {
    const int tt = t + k - PAD_;
    if (tt >= 0 && tt < T_)
      acc = fmaf(w0[k], a0[tt], fmaf(w1[k], a1[tt], acc));
  }
  loc[((size_t)b * NF_ + f) * T_ + t] = acc;
}

// ---------------------------------------------------------------------------
// Kernel 3 (fused): per (b, 16-t tile)
//   - pmem tile (16t x 128a, 8KB) staged to LDS via global_load_async_to_lds_b128
//   - ploc tile (16t x 128a) via WMMA (loc tile staged in LDS)
//   - energies[b][t] = sum_a v[a]*tanh(ploc + pq[b][a] + pmem), + mask
// 256 threads = 8 waves, wave j handles a-tile [16j, 16j+16).
// ---------------------------------------------------------------------------
__global__ __launch_bounds__(256) void energies_kernel(
    const float* __restrict__ loc, const float* __restrict__ Wloc,
    const float* __restrict__ pq, const float* __restrict__ pmem,
    const float* __restrict__ vw, const unsigned char* __restrict__ mask,
    float* __restrict__ energies) {
  __shared__ __align__(16) float lds_pmem[16 * A_];  // [t][a], 8KB
  __shared__ __align__(16) float lds_loc[16 * 34];   // [t][f], stride 34
  __shared__ float lds_e[8][16];
  const int b    = blockIdx.x;
  const int t0   = blockIdx.y * 16;
  const int tid  = threadIdx.x;
  const int wave = tid >> 5;
  const int lane = tid & 31;
  const int half = lane >> 4;
  const int idx  = lane & 15;

  // async-stage pmem[b][t0..t0+15][0..127] -> lds_pmem (2 rounds of b128/lane)
  {
    const int row  = tid >> 5;          // 0..7
    const int col4 = (tid & 31) * 4;    // 0,4,...,124
#pragma unroll
    for (int r = 0; r < 2; ++r) {
      const int rr = row + r * 8;
      async_load_b128((unsigned)(size_t)&lds_pmem[rr * A_ + col4],
                      pmem + ((size_t)b * T_ + t0 + rr) * A_ + col4);
    }
  }

  // stage loc[b][0..31][t0..t0+15] -> lds_loc[t][f]  (coalesced global reads)
  {
    const int f = tid >> 4;   // 0..15
    const int t = tid & 15;
    lds_loc[t * 34 + f]      = loc[((size_t)b * NF_ + f)      * T_ + t0 + t];
    lds_loc[t * 34 + f + 16] = loc[((size_t)b * NF_ + f + 16) * T_ + t0 + t];
  }
  wait_asynccnt0();
  __syncthreads();

  const int a0 = wave * 16;
  v8f c = {};
#pragma unroll
  for (int kk = 0; kk < 8; ++kk) {          // K = NF = 32, step 4
    const int kbase = 4 * kk + 2 * half;
    v2f a = *(const v2f*)(&lds_loc[idx * 34 + kbase]);                // A[m=idx][k]
    v2f bfr = *(const v2f*)(Wloc + (size_t)(a0 + idx) * NF_ + kbase); // B[k][n=idx]
    c = __builtin_amdgcn_wmma_f32_16x16x4_f32(false, a, false, bfr,
                                              (short)0, c, false, false);
  }

  // epilogue: D layout: c[i] = (M = i + 8*half, N = idx) of the 16x16 tile
  const int   acol = a0 + idx;
  const float pqv  = pq[(size_t)b * A_ + acol];
  const float vv   = vw[acol];
#pragma unroll
  for (int i = 0; i < 8; ++i) {
    const int m = i + 8 * half;
    const float x = c[i] + pqv + lds_pmem[m * A_ + acol];
    float val = tanhf(x) * vv;
#pragma unroll
    for (int off = 1; off < 16; off <<= 1)   // reduce over N within 16-lane group
      val += __shfl_xor(val, off, 32);
    if (idx == 0) lds_e[wave][m] = val;      // lane0 -> m=i, lane16 -> m=i+8
  }
  __syncthreads();

  if (tid < 16) {
    float e = 0.f;
#pragma unroll
    for (int w = 0; w < 8; ++w) e += lds_e[w][tid];   // reduce over a-tiles
    if (mask[(size_t)b * T_ + t0 + tid]) e = -__builtin_inff();
    energies[(size_t)b * T_ + t0 + tid] = e;
  }
}

// ---------------------------------------------------------------------------
// Kernel 4: softmax over T per batch row
// ---------------------------------------------------------------------------
__global__ __launch_bounds__(256) void softmax_kernel(
    const float* __restrict__ energies, float* __restrict__ weights) {
  __shared__ float red[8];
  const int b   = blockIdx.x;
  const int tid = threadIdx.x;
  const float* e = energies + (size_t)b * T_;

  float m = -__builtin_inff();
  for (int t = tid; t < T_; t += 256) m = fmaxf(m, e[t]);
#pragma unroll
  for (int off = 16; off > 0; off >>= 1) m = fmaxf(m, __shfl_xor(m, off, 32));
  if ((tid & 31) == 0) red[tid >> 5] = m;
  __syncthreads();
  float bm = red[0];
#pragma unroll
  for (int w = 1; w < 8; ++w) bm = fmaxf(bm, red[w]);
  __syncthreads();

  float s = 0.f;
  for (int t = tid; t < T_; t += 256) s += __expf(e[t] - bm);
#pragma unroll
  for (int off = 16; off > 0; off >>= 1) s += __shfl_xor(s, off, 32);
  if ((tid & 31) == 0) red[tid >> 5] = s;
  __syncthreads();
  float tot = 0.f;
#pragma unroll
  for (int w = 0; w < 8; ++w) tot += red[w];
  const float inv = 1.f / tot;

  for (int t = tid; t < T_; t += 256)
    weights[(size_t)b * T_ + t] = __expf(e[t] - bm) * inv;
}

// ---------------------------------------------------------------------------
// Kernel 5: zero the context region of d_out (atomically accumulated below)
// ---------------------------------------------------------------------------
__global__ void zero_kernel(float* __restrict__ p, int n) {
  const int i = blockIdx.x * 256 + threadIdx.x;
  if (i < n) p[i] = 0.f;
}

// ---------------------------------------------------------------------------
// Kernel 6: context[b][e] = sum_t w[b][t] * memory[b][t][e]
// grid (B, 16 t-segments); 128 threads, each owns 4 consecutive e (float4 ->
// global_load_b128, 2KB fully-coalesced per t); weights chunk in LDS;
// partial sums via global f32 atomics.
// ---------------------------------------------------------------------------
__global__ __launch_bounds__(128) void context_kernel(
    const float* __restrict__ weights, const float* __restrict__ memory,
    float* __restrict__ ctx) {
  __shared__ float wsh[128];
  const int b    = blockIdx.x;
  const int e4   = threadIdx.x * 4;          // 0..508
  const int tbeg = blockIdx.y * (T_ / 16);   // 125 t-values per segment
  const int tend = tbeg + (T_ / 16);
  float ax = 0.f, ay = 0.f, az = 0.f, aw_ = 0.f;
  for (int tc = tbeg; tc < tend; tc += 128) {
    int nt = tend - tc; if (nt > 128) nt = 128;
    __syncthreads();
    if ((int)threadIdx.x < nt)
      wsh[threadIdx.x] = weights[(size_t)b * T_ + tc + threadIdx.x];
    __syncthreads();
    for (int i = 0; i < nt; ++i) {
      const float4 mv =
          *(const float4*)&memory[((size_t)b * T_ + tc + i) * E_ + e4];
      const float w = wsh[i];
      ax = fmaf(w, mv.x, ax);
      ay = fmaf(w, mv.y, ay);
      az = fmaf(w, mv.z, az);
      aw_ = fmaf(w, mv.w, aw_);
    }
  }
  float* dst = &ctx[(size_t)b * E_ + e4];
  atomicAdd(dst + 0, ax);
  atomicAdd(dst + 1, ay);
  atomicAdd(dst + 2, az);
  atomicAdd(dst + 3, aw_);
}

// ---------------------------------------------------------------------------
extern "C" void kernel_launch(void* const* d_in, const int* in_sizes, int n_in,
                              void* d_out, int out_size, void* d_ws, size_t ws_size,
                              hipStream_t stream) {
  const float*         query = (const float*)d_in[0];
  const float*         mem   = (const float*)d_in[1];
  const float*         pmem  = (const float*)d_in[2];
  const float*         awcat = (const float*)d_in[3];
  const unsigned char* mask  = (const unsigned char*)d_in[4];
  const float*         Wq    = (const float*)d_in[5];
  const float*         convw = (const float*)d_in[6];
  const float*         Wloc  = (const float*)d_in[7];
  const float*         vw    = (const float*)d_in[8];

  float* out = (float*)d_out;
  float* ctx = out;                   // (B, E)  = 32768 floats
  float* wts = out + B_ * E_;         // (B, T)  = 128000 floats

  float* ws       = (float*)d_ws;
  float* pq       = ws;                              // 64*128      = 8192
  float* loc      = pq + B_ * A_;                    // 64*32*2000  = 4,096,000
  float* energies = loc + (size_t)B_ * NF_ * T_;     // 64*2000     = 128,000

  pq_wmma_kernel<<<dim3(B_ / 16, A_ / 16), 32, 0, stream>>>(query, Wq, pq);
  conv_kernel<<<dim3(B_, NF_, (T_ + 255) / 256), 256, 0, stream>>>(awcat, convw, loc);
  energies_kernel<<<dim3(B_, T_ / 16), 256, 0, stream>>>(loc, Wloc, pq, pmem, vw,
                                                         mask, energies);
  softmax_kernel<<<B_, 256, 0, stream>>>(energies, wts);
  zero_kernel<<<(B_ * E_ + 255) / 256, 256, 0, stream>>>(ctx, B_ * E_);
  context_kernel<<<dim3(B_, 16), 128, 0, stream>>>(wts, mem, ctx);
}